// GCN_audio_top_med_fea_37434934952165
// MI455X (gfx1250) — compile-verified
//
#include <hip/hip_runtime.h>
#include <hip/hip_bf16.h>
#include <math.h>

typedef __attribute__((ext_vector_type(2))) float v2f;
typedef __attribute__((ext_vector_type(4))) float v4f;
typedef __attribute__((ext_vector_type(8))) float v8f;

#define NBATCH 64
#define CCH    1024   // channels
#define HWD    1024   // W*H
#define BN_INV 0.99999500003749937f   // 1/sqrt(1+1e-5)

// ---------------------------------------------------------------------------
// Kernel 1a: channel-quarter partial sums of sounds, b128 non-temporal loads.
// partial[cq][n][hw] = sum_{c in quarter cq} sounds[n,c,hw]
// ---------------------------------------------------------------------------
__global__ __launch_bounds__(256) void reduce_partial_kernel(
    const float* __restrict__ sounds, float* __restrict__ partial) {
  const int n   = blockIdx.x >> 2;
  const int cq  = blockIdx.x & 3;
  const int hw4 = threadIdx.x << 2;
  const float* p = sounds + (size_t)n * CCH * HWD + (size_t)cq * 256 * HWD + hw4;
  v4f a0 = {}, a1 = {}, a2 = {}, a3 = {};
  #pragma unroll 2
  for (int c = 0; c < 256; c += 4) {
    a0 += __builtin_nontemporal_load((const v4f*)(p + (size_t)(c + 0) * HWD));
    a1 += __builtin_nontemporal_load((const v4f*)(p + (size_t)(c + 1) * HWD));
    a2 += __builtin_nontemporal_load((const v4f*)(p + (size_t)(c + 2) * HWD));
    a3 += __builtin_nontemporal_load((const v4f*)(p + (size_t)(c + 3) * HWD));
  }
  v4f r = (a0 + a1) + (a2 + a3);
  *(v4f*)(partial + ((size_t)cq * NBATCH + n) * HWD + hw4) = r;
}

// Kernel 1b: feat[n,hw] = sum_cq partial[cq][n][hw]
__global__ __launch_bounds__(256) void reduce_final_kernel(
    const float* __restrict__ partial, float* __restrict__ feat) {
  const int n   = blockIdx.x;
  const int hw4 = threadIdx.x << 2;
  v4f s0 = *(const v4f*)(partial + ((size_t)0 * NBATCH + n) * HWD + hw4);
  v4f s1 = *(const v4f*)(partial + ((size_t)1 * NBATCH + n) * HWD + hw4);
  v4f s2 = *(const v4f*)(partial + ((size_t)2 * NBATCH + n) * HWD + hw4);
  v4f s3 = *(const v4f*)(partial + ((size_t)3 * NBATCH + n) * HWD + hw4);
  *(v4f*)(feat + (size_t)n * HWD + hw4) = (s0 + s1) + (s2 + s3);
}

// ---------------------------------------------------------------------------
// Kernel 2: per-row bitonic argsort (desc by value, ties -> lower index first,
// matching stable argsort(-feat)). Emits scene indices + rows/cols outputs.
// ---------------------------------------------------------------------------
__global__ __launch_bounds__(512) void sort_kernel(const float* __restrict__ feat,
                                                   int* __restrict__ scene,
                                                   float* __restrict__ rowsOut,
                                                   float* __restrict__ colsOut) {
  __shared__ float v[1024];
  __shared__ int   ix[1024];
  const int n = blockIdx.x;
  const int tid = threadIdx.x;
  v[tid]       = feat[n * 1024 + tid];       ix[tid]       = tid;
  v[tid + 512] = feat[n * 1024 + tid + 512]; ix[tid + 512] = tid + 512;
  __syncthreads();
  for (int k = 2; k <= 1024; k <<= 1) {
    for (int j = k >> 1; j > 0; j >>= 1) {
      #pragma unroll
      for (int half = 0; half < 2; ++half) {
        const int i = tid + half * 512;
        const int l = i ^ j;
        if (l > i) {
          const bool asc = ((i & k) == 0);
          const bool pre = (v[i] > v[l]) || ((v[i] == v[l]) && (ix[i] < ix[l]));
          if (asc ? !pre : pre) {
            float tv = v[i]; v[i] = v[l]; v[l] = tv;
            int   tx = ix[i]; ix[i] = ix[l]; ix[l] = tx;
          }
        }
      }
      __syncthreads();
    }
  }
  int pos = -1;
  if (tid < 16)                     pos = tid;               // top-K
  else if (tid >= 503 && tid < 519) pos = 16 + (tid - 503);  // med-K (start=503)
  if (pos >= 0) {
    const int s = ix[tid];
    scene[n * 32 + pos]   = s;
    rowsOut[n * 32 + pos] = (float)(s >> 5);
    colsOut[n * 32 + pos] = (float)(s & 31);
  }
}

// ---------------------------------------------------------------------------
// Kernel 3: per-batch head GEMM (32x1024 @ 1024x256) + BN + ReLU, then the
// two 16x16 Laplacian mixes, all via V_WMMA_F32_16X16X4_F32. One block per n.
// ---------------------------------------------------------------------------
__global__ __launch_bounds__(256) void head_mix_kernel(
    const float* __restrict__ sounds, const float* __restrict__ conv1_w,
    const float* __restrict__ conv1_b, const float* __restrict__ cbn_g,
    const float* __restrict__ cbn_b, const int* __restrict__ scene,
    float* __restrict__ fus) {
  __shared__ float xl[32][257];       // head output (nodes x 256 feats), padded
  __shared__ float Adj[2][16][17];
  __shared__ float Lm[2][16][17];
  __shared__ float dinv[32];

  const int n    = blockIdx.x;
  const int tid  = threadIdx.x;
  const int wave = tid >> 5;
  const int lane = tid & 31;
  const int hi   = lane >> 4;
  const int lm   = lane & 15;

  // -------- head GEMM: wave -> (mtile, 4 ntiles) --------
  const int mtile = wave & 1;
  const int ntb   = (wave >> 1) << 2;
  const int s     = scene[n * 32 + mtile * 16 + lm];
  const float* ap = sounds + (size_t)n * CCH * HWD + s;  // + k*HWD walks channels

  v8f acc0 = {}, acc1 = {}, acc2 = {}, acc3 = {};
  for (int kb = 0; kb < CCH; kb += 4) {
    const int k0 = kb + 2 * hi;
    v2f a;
    a.x = ap[(size_t)k0 * HWD];
    a.y = ap[(size_t)(k0 + 1) * HWD];
    v2f b0 = *(const v2f*)(conv1_w + (size_t)((ntb + 0) * 16 + lm) * CCH + k0);
    v2f b1 = *(const v2f*)(conv1_w + (size_t)((ntb + 1) * 16 + lm) * CCH + k0);
    v2f b2 = *(const v2f*)(conv1_w + (size_t)((ntb + 2) * 16 + lm) * CCH + k0);
    v2f b3 = *(const v2f*)(conv1_w + (size_t)((ntb + 3) * 16 + lm) * CCH + k0);
    acc0 = __builtin_amdgcn_wmma_f32_16x16x4_f32(false, a, false, b0, (short)0, acc0, false, false);
    acc1 = __builtin_amdgcn_wmma_f32_16x16x4_f32(false, a, false, b1, (short)0, acc1, false, false);
    acc2 = __builtin_amdgcn_wmma_f32_16x16x4_f32(false, a, false, b2, (short)0, acc2, false, false);
    acc3 = __builtin_amdgcn_wmma_f32_16x16x4_f32(false, a, false, b3, (short)0, acc3, false, false);
  }
  #pragma unroll
  for (int t = 0; t < 4; ++t) {
    v8f av = (t == 0) ? acc0 : (t == 1) ? acc1 : (t == 2) ? acc2 : acc3;
    const int col = (ntb + t) * 16 + lm;
    const float gg = cbn_g[col], bb = cbn_b[col], bs = conv1_b[col];
    #pragma unroll
    for (int r = 0; r < 8; ++r) {
      const int row = mtile * 16 + r + 8 * hi;
      float vv = gg * ((av[r] + bs) * BN_INV) + bb;
      xl[row][col] = fmaxf(vv, 0.0f);
    }
  }
  __syncthreads();

  // -------- symmetric-normalized adjacency for top/med --------
  {
    const int i = tid >> 4, j = tid & 15;
    #pragma unroll
    for (int g = 0; g < 2; ++g) {
      const int si = scene[n * 32 + g * 16 + i];
      const int sj = scene[n * 32 + g * 16 + j];
      const float dr = (float)(si >> 5) - (float)(sj >> 5);
      const float dc = (float)(si & 31) - (float)(sj & 31);
      Adj[g][i][j] = 1.0f / (1.0f + sqrtf(dr * dr + dc * dc));
    }
  }
  __syncthreads();
  if (tid < 32) {
    const int g = tid >> 4, i = tid & 15;
    float sum = 0.f;
    #pragma unroll
    for (int j = 0; j < 16; ++j) sum += Adj[g][i][j];
    dinv[tid] = 1.0f / sqrtf(sum);
  }
  __syncthreads();
  {
    const int i = tid >> 4, j = tid & 15;
    #pragma unroll
    for (int g = 0; g < 2; ++g)
      Lm[g][i][j] = Adj[g][i][j] * dinv[g * 16 + i] * dinv[g * 16 + j];
  }
  __syncthreads();

  // -------- mix: fus = 0.5*(Ltop@x + Lmed@x_m)  (16x16 @ 16x256) --------
  #pragma unroll
  for (int tt = 0; tt < 2; ++tt) {
    const int t = wave * 2 + tt;
    const int col = t * 16 + lm;
    v8f at = {}, am = {};
    #pragma unroll
    for (int kb = 0; kb < 16; kb += 4) {
      const int k0 = kb + 2 * hi;
      v2f aT, aM, bT, bM;
      aT.x = Lm[0][lm][k0];      aT.y = Lm[0][lm][k0 + 1];
      aM.x = Lm[1][lm][k0];      aM.y = Lm[1][lm][k0 + 1];
      bT.x = xl[k0][col];        bT.y = xl[k0 + 1][col];
      bM.x = xl[16 + k0][col];   bM.y = xl[16 + k0 + 1][col];
      at = __builtin_amdgcn_wmma_f32_16x16x4_f32(false, aT, false, bT, (short)0, at, false, false);
      am = __builtin_amdgcn_wmma_f32_16x16x4_f32(false, aM, false, bM, (short)0, am, false, false);
    }
    #pragma unroll
    for (int r = 0; r < 8; ++r) {
      const int row = r + 8 * hi;
      fus[(size_t)n * 4096 + row * 256 + col] = 0.5f * (at[r] + am[r]);
    }
  }
}

// ---------------------------------------------------------------------------
// Kernel 4/5: fused FC + bias + BN + ReLU.  M=64 fixed; one 16x16 tile/wave.
// ---------------------------------------------------------------------------
__global__ __launch_bounds__(256) void fc_kernel(
    const float* __restrict__ A, const float* __restrict__ W,
    const float* __restrict__ bias, const float* __restrict__ gam,
    const float* __restrict__ bet, float* __restrict__ out,
    const int Kdim, const int Ndim) {
  const int wave = threadIdx.x >> 5;
  const int lane = threadIdx.x & 31;
  const int hi = lane >> 4;
  const int lm = lane & 15;
  const int gw = blockIdx.x * 8 + wave;
  const int mtile = gw & 3;          // 4 row tiles (M=64)
  const int ntile = gw >> 2;
  const int m   = mtile * 16 + lm;
  const int col = ntile * 16 + lm;
  const float* ap = A + (size_t)m * Kdim;
  const float* wp = W + (size_t)col * Kdim;
  v8f acc0 = {}, acc1 = {};
  for (int kb = 0; kb < Kdim; kb += 8) {
    const int k0 = kb + 2 * hi;
    v2f a0 = *(const v2f*)(ap + k0);
    v2f b0 = *(const v2f*)(wp + k0);
    v2f a1 = *(const v2f*)(ap + k0 + 4);
    v2f b1 = *(const v2f*)(wp + k0 + 4);
    acc0 = __builtin_amdgcn_wmma_f32_16x16x4_f32(false, a0, false, b0, (short)0, acc0, false, false);
    acc1 = __builtin_amdgcn_wmma_f32_16x16x4_f32(false, a1, false, b1, (short)0, acc1, false, false);
  }
  v8f acc = acc0 + acc1;
  const float gg = gam[col], bb = bet[col], bs = bias[col];
  #pragma unroll
  for (int r = 0; r < 8; ++r) {
    const int row = mtile * 16 + r + 8 * hi;
    float vv = gg * ((acc[r] + bs) * BN_INV) + bb;
    out[(size_t)row * Ndim + col] = fmaxf(vv, 0.0f);
  }
}

// ---------------------------------------------------------------------------
extern "C" void kernel_launch(void* const* d_in, const int* in_sizes, int n_in,
                              void* d_out, int out_size, void* d_ws, size_t ws_size,
                              hipStream_t stream) {
  (void)in_sizes; (void)n_in; (void)out_size; (void)ws_size;
  const float* sounds  = (const float*)d_in[0];
  const float* conv1_w = (const float*)d_in[1];
  const float* conv1_b = (const float*)d_in[2];
  const float* cbn_g   = (const float*)d_in[3];
  const float* cbn_b   = (const float*)d_in[4];
  const float* fc1_w   = (const float*)d_in[5];
  const float* fc1_b   = (const float*)d_in[6];
  const float* bn1_g   = (const float*)d_in[7];
  const float* bn1_b   = (const float*)d_in[8];
  const float* fc2_w   = (const float*)d_in[9];
  const float* fc2_b   = (const float*)d_in[10];
  const float* bn2_g   = (const float*)d_in[11];
  const float* bn2_b   = (const float*)d_in[12];

  float* out  = (float*)d_out;                // [64*512] o, [64*32] rows, [64*32] cols
  float* feat = (float*)d_ws;                 // 64*1024
  float* fus  = feat + 64 * 1024;             // 64*4096 (also reused as partial[4][64][1024])
  float* o1   = fus + 64 * 4096;              // 64*1024
  int* scene  = (int*)(o1 + 64 * 1024);       // 64*32

  float* partial = fus;                       // reuse: dead before head_mix writes fus

  float* rowsOut = out + 64 * 512;
  float* colsOut = rowsOut + 64 * 32;

  reduce_partial_kernel<<<256, 256, 0, stream>>>(sounds, partial);
  reduce_final_kernel<<<64, 256, 0, stream>>>(partial, feat);
  sort_kernel<<<64, 512, 0, stream>>>(feat, scene, rowsOut, colsOut);
  head_mix_kernel<<<64, 256, 0, stream>>>(sounds, conv1_w, conv1_b, cbn_g, cbn_b, scene, fus);
  fc_kernel<<<32, 256, 0, stream>>>(fus, fc1_w, fc1_b, bn1_g, bn1_b, o1, 4096, 1024);
  fc_kernel<<<16, 256, 0, stream>>>(o1, fc2_w, fc2_b, bn2_g, bn2_b, out, 1024, 512);
}